// MultiBoxLoss_44281112821988
// MI455X (gfx1250) — compile-verified
//
#include <hip/hip_runtime.h>
#include <math.h>

// ---- constants from the reference ----
#define F_ALPHA_C 0.25f
#define L_ALPHA_C 0.5f
#define L_GAMMA_C 1.5f
#define L_BETA_C  0.11f
// b = exp(L_GAMMA/L_ALPHA) - 1 = exp(3) - 1
#define L_B_C 19.085536923187668f

#define LOG2E_C 1.44269504088896340736f
#define LN2_C   0.69314718055994530942f

typedef unsigned int uint32x4_t_ __attribute__((ext_vector_type(4)));
typedef int          int32x8_t_  __attribute__((ext_vector_type(8)));
typedef int          int32x4_t_  __attribute__((ext_vector_type(4)));

struct Accum {
    double l1_sum;
    double fl_sum;
    unsigned int pos_cnt;
    unsigned int ign_cnt;
};

// ---- fast hardware transcendentals (raw v_exp_f32 / v_log_f32 / v_rcp_f32) ----
__device__ __forceinline__ float fast_exp2(float x) {
#if __has_builtin(__builtin_amdgcn_exp2f)
    return __builtin_amdgcn_exp2f(x);
#else
    return exp2f(x);
#endif
}
__device__ __forceinline__ float fast_log2(float x) {   // only called with x >= 1
#if __has_builtin(__builtin_amdgcn_logf)
    return __builtin_amdgcn_logf(x);
#else
    return log2f(x);
#endif
}
__device__ __forceinline__ float fast_rcp(float x) {    // only called with x >= 1
#if __has_builtin(__builtin_amdgcn_rcpf)
    return __builtin_amdgcn_rcpf(x);
#else
    return 1.0f / x;
#endif
}

__device__ __forceinline__ float iou_pt(float ax0, float ay0, float ax1, float ay1,
                                        float bx0, float by0, float bx1, float by1) {
    float lx = fmaxf(ax0, bx0), ly = fmaxf(ay0, by0);
    float rx = fminf(ax1, bx1), ry = fminf(ay1, by1);
    float w = fmaxf(rx - lx, 0.0f), h = fmaxf(ry - ly, 0.0f);
    float inter = w * h;
    float aa = (ax1 - ax0) * (ay1 - ay0);
    float ab = (bx1 - bx0) * (by1 - by0);
    return inter / (aa + ab - inter);
}

__device__ __forceinline__ float balanced_l1(float diff) {
    float d = fabsf(diff);
    if (d < L_BETA_C) {
        return (L_ALPHA_C / L_B_C) * (L_B_C * d + 1.0f) * logf(L_B_C * d / L_BETA_C + 1.0f)
               - L_ALPHA_C * d;
    }
    return L_GAMMA_C * d + (L_GAMMA_C / L_B_C - L_ALPHA_C * L_BETA_C);
}

// Focal-loss contribution of one logit given (tpos, maskoff).
// t==1 path: ce=-log_sigmoid(x)=softplus(-x), pt=sigmoid(-x), w=alpha*pt
// t==0 path: ce=-log_sigmoid(-x)=softplus(x), pt=sigmoid(x),  w=(1-alpha)*pt
// Unified with y = tpos ? -x : x, z = e^y:  pt = z/(1+z), ce = ln(1+z).
__device__ __forceinline__ float focal_elem(float x, bool tpos, bool maskoff) {
    float y = tpos ? -x : x;
    y = fminf(fmaxf(y, -60.0f), 60.0f);
    float z  = fast_exp2(y * LOG2E_C);        // v_exp_f32
    float zp = 1.0f + z;                      // zp >= 1
    float pt = z * fast_rcp(zp);              // v_rcp_f32
    float ce = fast_log2(zp) * LN2_C;         // v_log_f32
    float w  = (tpos ? F_ALPHA_C : (1.0f - F_ALPHA_C)) * pt;
    return maskoff ? 0.0f : w * ce;
}

__device__ __forceinline__ float wave_red_f(float v) {
    #pragma unroll
    for (int off = 16; off > 0; off >>= 1) v += __shfl_down(v, off, 32);
    return v;
}
__device__ __forceinline__ unsigned int wave_red_u(unsigned int v) {
    #pragma unroll
    for (int off = 16; off > 0; off >>= 1) v += __shfl_down(v, off, 32);
    return v;
}

// ------------------------------------------------------------------
__global__ void init_accum_kernel(Accum* a) {
    a->l1_sum = 0.0; a->fl_sum = 0.0; a->pos_cnt = 0u; a->ign_cnt = 0u;
}

// ------------------------------------------------------------------
// One block per (b,t): argmax over priors (tie -> smallest prior index).
__global__ void best_prior_kernel(const float* __restrict__ targets,
                                  const float* __restrict__ priors,
                                  int* __restrict__ bp,
                                  int B, int T, int P) {
    int bt = blockIdx.x;
    const float* tr = targets + (size_t)bt * 5;
    float tx0 = tr[0], ty0 = tr[1], tx1 = tr[2], ty1 = tr[3];
    float bestv = -1.0f; int besti = 0;
    for (int p = threadIdx.x; p < P; p += blockDim.x) {
        const float* pp = priors + (size_t)p * 4;
        float cx = pp[0], cy = pp[1], w = pp[2], h = pp[3];
        float v = iou_pt(tx0, ty0, tx1, ty1,
                         cx - 0.5f * w, cy - 0.5f * h, cx + 0.5f * w, cy + 0.5f * h);
        if (v > bestv) { bestv = v; besti = p; }
    }
    __shared__ float rv[256];
    __shared__ int   ri[256];
    rv[threadIdx.x] = bestv; ri[threadIdx.x] = besti;
    __syncthreads();
    for (int s = blockDim.x >> 1; s > 0; s >>= 1) {
        if ((int)threadIdx.x < s) {
            float v2 = rv[threadIdx.x + s]; int i2 = ri[threadIdx.x + s];
            if (v2 > rv[threadIdx.x] ||
                (v2 == rv[threadIdx.x] && i2 < ri[threadIdx.x])) {
                rv[threadIdx.x] = v2; ri[threadIdx.x] = i2;
            }
        }
        __syncthreads();
    }
    if (threadIdx.x == 0) bp[bt] = ri[0];
}

// ------------------------------------------------------------------
// Per (b,p): matching + loc loss. Truths staged into LDS via TDM.
__global__ void match_loc_kernel(const float* __restrict__ targets,
                                 const float* __restrict__ priors,
                                 const float* __restrict__ loc_data,
                                 const int* __restrict__ bp,
                                 int* __restrict__ clspk,
                                 Accum* acc,
                                 int B, int T, int P) {
    __shared__ float sh_t[5 * 32];      // T<=32 rows of [x0,y0,x1,y1,label]
    __shared__ int   sh_bp[32];
    __shared__ float wsum[8];
    __shared__ unsigned int wcnt[8];

    int b = blockIdx.y;

#if __has_builtin(__builtin_amdgcn_tensor_load_to_lds) && __has_builtin(__builtin_amdgcn_s_wait_tensorcnt)
    if (threadIdx.x < 32u) {
        unsigned long long g = (unsigned long long)(size_t)(const void*)(targets + (size_t)b * T * 5);
        // For the LDS aperture, low 32 bits of the generic address == LDS byte offset.
        unsigned int lds = (unsigned int)(size_t)(void*)&sh_t[0];
        unsigned int n = (unsigned int)(T * 5);   // elements (4B each), 1D tile
        uint32x4_t_ g0;
        g0[0] = 1u;                                     // count=1 (valid), user mode
        g0[1] = lds;                                    // lds_addr
        g0[2] = (unsigned int)g;                        // global_addr[31:0]
        g0[3] = ((unsigned int)(g >> 32) & 0x01FFFFFFu) // global_addr[56:32]
                | (2u << 30);                           // type=2 ("image")
        int32x8_t_ g1;
        g1[0] = (int)(2u << 16);                        // wg_mask=0, data_size=2 (4B)
        g1[1] = (int)((n & 0xFFFFu) << 16);             // tensor_dim0[15:0]
        g1[2] = (int)((n >> 16) | (1u << 16));          // tensor_dim0[31:16] | tensor_dim1=1
        g1[3] = (int)((n & 0xFFFFu) << 16);             // tensor_dim1[31:16]=0 | tile_dim0=n
        g1[4] = 1;                                      // tile_dim1=1, tile_dim2=0
        g1[5] = (int)n;                                 // tensor_dim0_stride
        g1[6] = 0;
        g1[7] = 0;
        int32x4_t_ z4 = {0, 0, 0, 0};
        int32x8_t_ z8 = {0, 0, 0, 0, 0, 0, 0, 0};
        __builtin_amdgcn_tensor_load_to_lds(g0, g1, z4, z4, z8, 0);
        __builtin_amdgcn_s_wait_tensorcnt(0);
    }
#else
    for (int i = threadIdx.x; i < T * 5; i += blockDim.x)
        sh_t[i] = targets[(size_t)b * T * 5 + i];
#endif
    if ((int)threadIdx.x < T) sh_bp[threadIdx.x] = bp[b * T + threadIdx.x];
    __syncthreads();

    int p = blockIdx.x * blockDim.x + threadIdx.x;
    float l1s = 0.0f;
    unsigned int cnt = 0u;   // pos in [15:0], ign in [31:16]
    if (p < P) {
        const float* pp = priors + (size_t)p * 4;
        float cx = pp[0], cy = pp[1], pw = pp[2], ph = pp[3];
        float px0 = cx - 0.5f * pw, py0 = cy - 0.5f * ph;
        float px1 = cx + 0.5f * pw, py1 = cy + 0.5f * ph;

        float bestv = -1.0f; int bt_ = 0;
        for (int t = 0; t < T; ++t) {              // first-max => smallest t on ties
            const float* tr = &sh_t[t * 5];
            float v = iou_pt(tr[0], tr[1], tr[2], tr[3], px0, py0, px1, py1);
            if (v > bestv) { bestv = v; bt_ = t; }
        }
        for (int t = 0; t < T; ++t)                // .at[].set override, last t wins
            if (sh_bp[t] == p) { bestv = 2.0f; bt_ = t; }

        bool pos = bestv >= 0.5f;
        bool ign = (!pos) && (bestv >= 0.4f);
        bool neg = bestv < 0.4f;

        const float* tr = &sh_t[bt_ * 5];
        int cls = neg ? 0 : ((int)tr[4] + 1);
        clspk[(size_t)b * P + p] = cls | (ign ? 0x100 : 0);

        if (pos) {
            float mx0 = tr[0], my0 = tr[1], mx1 = tr[2], my1 = tr[3];
            float gx = ((mx0 + mx1) * 0.5f - cx) / (0.1f * pw);
            float gy = ((my0 + my1) * 0.5f - cy) / (0.1f * ph);
            float gw = logf((mx1 - mx0) / pw) / 0.2f;
            float gh = logf((my1 - my0) / ph) / 0.2f;
            const float* ld = loc_data + ((size_t)b * P + p) * 4;
            l1s = balanced_l1(ld[0] - gx) + balanced_l1(ld[1] - gy)
                + balanced_l1(ld[2] - gw) + balanced_l1(ld[3] - gh);
            cnt = 1u;
        }
        if (ign) cnt |= (1u << 16);
    }

    // wave32 reduction, then cross-wave via LDS (256 threads = 8 waves)
    float s = wave_red_f(l1s);
    unsigned int c = wave_red_u(cnt);
    int lane = threadIdx.x & 31, wid = threadIdx.x >> 5;
    if (lane == 0) { wsum[wid] = s; wcnt[wid] = c; }
    __syncthreads();
    if (wid == 0) {
        s = (lane < 8) ? wsum[lane] : 0.0f;
        c = (lane < 8) ? wcnt[lane] : 0u;
        s = wave_red_f(s);
        c = wave_red_u(c);
        if (lane == 0) {
            atomicAdd(&acc->l1_sum, (double)s);
            atomicAdd(&acc->pos_cnt, c & 0xFFFFu);
            atomicAdd(&acc->ign_cnt, c >> 16);
        }
    }
}

// ------------------------------------------------------------------
// The 84 MB stream, float4 per thread (requires C % 4 == 0).
__global__ void focal_kernel_v4(const float4* __restrict__ conf4,
                                const int* __restrict__ clspk,
                                Accum* acc,
                                unsigned int N4, int Cq) {
    unsigned int i = blockIdx.x * blockDim.x + threadIdx.x;
    float fl = 0.0f;
    if (i < N4) {
        // Speculative prefetch ~1 MB ahead of the stream (global_prefetch_b8).
        __builtin_prefetch((const char*)(conf4 + i) + (1u << 20), 0, 0);
        float4 x4 = conf4[i];
        unsigned int row = i / (unsigned int)Cq;
        int c0 = (int)(i - row * (unsigned int)Cq) * 4;   // class of lane .x
        int pk = clspk[row];
        int cls = pk & 0xFF;
        bool ign = (pk & 0x100) != 0;
        float xs[4] = {x4.x, x4.y, x4.z, x4.w};
        #pragma unroll
        for (int k = 0; k < 4; ++k) {
            bool onehot  = (cls == c0 + k + 1);
            bool tpos    = onehot && !ign;
            bool maskoff = onehot && ign;
            fl += focal_elem(xs[k], tpos, maskoff);
        }
    }
    float s = wave_red_f(fl);
    __shared__ float wsum[16];
    int lane = threadIdx.x & 31, wid = threadIdx.x >> 5;
    if (lane == 0) wsum[wid] = s;
    __syncthreads();
    if (wid == 0) {
        s = (lane < 16) ? wsum[lane] : 0.0f;
        s = wave_red_f(s);
        if (lane == 0) atomicAdd(&acc->fl_sum, (double)s);
    }
}

// Scalar fallback for C % 4 != 0.
__global__ void focal_kernel_s(const float* __restrict__ conf,
                               const int* __restrict__ clspk,
                               Accum* acc,
                               unsigned int N, int C) {
    unsigned int i = blockIdx.x * blockDim.x + threadIdx.x;
    float fl = 0.0f;
    if (i < N) {
        __builtin_prefetch(conf + i + (1u << 18), 0, 0);
        float x = conf[i];
        unsigned int row = i / (unsigned int)C;
        int c = (int)(i - row * (unsigned int)C);
        int pk = clspk[row];
        int cls = pk & 0xFF;
        bool ign = (pk & 0x100) != 0;
        bool onehot = (cls == c + 1);
        fl = focal_elem(x, onehot && !ign, onehot && ign);
    }
    float s = wave_red_f(fl);
    __shared__ float wsum[16];
    int lane = threadIdx.x & 31, wid = threadIdx.x >> 5;
    if (lane == 0) wsum[wid] = s;
    __syncthreads();
    if (wid == 0) {
        s = (lane < 16) ? wsum[lane] : 0.0f;
        s = wave_red_f(s);
        if (lane == 0) atomicAdd(&acc->fl_sum, (double)s);
    }
}

// ------------------------------------------------------------------
__global__ void finalize_kernel(const Accum* __restrict__ acc,
                                float* __restrict__ out,
                                unsigned long long NPC) {
    unsigned int npos = acc->pos_cnt;
    if (npos < 1u) npos = 1u;
    unsigned long long mask_cnt = NPC - (unsigned long long)acc->ign_cnt;
    if (mask_cnt < 1ull) mask_cnt = 1ull;
    out[0] = (float)(acc->l1_sum / (4.0 * (double)npos));
    out[1] = (float)(acc->fl_sum / (double)mask_cnt);
}

// ------------------------------------------------------------------
extern "C" void kernel_launch(void* const* d_in, const int* in_sizes, int n_in,
                              void* d_out, int out_size, void* d_ws, size_t ws_size,
                              hipStream_t stream) {
    const float* loc_data  = (const float*)d_in[0];
    const float* conf_data = (const float*)d_in[1];
    const float* priors    = (const float*)d_in[2];
    const float* targets   = (const float*)d_in[3];

    int P = in_sizes[2] / 4;
    int B = in_sizes[0] / (P * 4);
    int C = in_sizes[1] / (B * P);
    int T = in_sizes[3] / (B * 5);

    // workspace layout
    char* ws = (char*)d_ws;
    Accum* acc  = (Accum*)ws;                       // 24 B
    int*   bp   = (int*)(ws + 256);                 // B*T ints
    size_t bp_bytes = ((size_t)B * T * sizeof(int) + 255) & ~(size_t)255;
    int*   clspk = (int*)(ws + 256 + bp_bytes);     // B*P ints

    init_accum_kernel<<<1, 1, 0, stream>>>(acc);

    best_prior_kernel<<<dim3(B * T), dim3(256), 0, stream>>>(
        targets, priors, bp, B, T, P);

    match_loc_kernel<<<dim3((P + 255) / 256, B), dim3(256), 0, stream>>>(
        targets, priors, loc_data, bp, clspk, acc, B, T, P);

    unsigned int N = (unsigned int)B * (unsigned int)P * (unsigned int)C;
    if ((C & 3) == 0) {
        unsigned int N4 = N / 4u;
        focal_kernel_v4<<<dim3((N4 + 511) / 512), dim3(512), 0, stream>>>(
            (const float4*)conf_data, clspk, acc, N4, C / 4);
    } else {
        focal_kernel_s<<<dim3((N + 511) / 512), dim3(512), 0, stream>>>(
            conf_data, clspk, acc, N, C);
    }

    finalize_kernel<<<1, 1, 0, stream>>>(
        acc, (float*)d_out, (unsigned long long)B * P * C);
}